// AttentionLayer_23278722744987
// MI455X (gfx1250) — compile-verified
//
#include <hip/hip_runtime.h>
#include <hip/hip_bf16.h>
#include <math.h>

typedef float v2f __attribute__((ext_vector_type(2)));
typedef float v8f __attribute__((ext_vector_type(8)));
typedef int   v4i __attribute__((ext_vector_type(4)));
typedef __attribute__((address_space(3))) v4i* lds_v4i_ptr;

#define DSIM 256
#define FATT 64
#define SLOPE 0.2f

#if defined(__has_builtin)
#if __has_builtin(__builtin_amdgcn_global_load_async_to_lds_b128) && \
    __has_builtin(__builtin_amdgcn_s_wait_asynccnt)
#define USE_ASYNC_LDS 1
#endif
#endif

// ---------------------------------------------------------------------------
// Init scratch: e_max = -inf, denom = 0, hacc = 0
// ---------------------------------------------------------------------------
__global__ __launch_bounds__(256)
void init_bufs(float* __restrict__ e_max, float* __restrict__ denom,
               float* __restrict__ hacc, int n_nodes) {
    int i = blockIdx.x * blockDim.x + threadIdx.x;
    int total = n_nodes * FATT;
    if (i < total) hacc[i] = 0.0f;
    if (i < n_nodes) {
        e_max[i] = __int_as_float(0xFF800000);   // -inf
        denom[i] = 0.0f;
    }
}

// ---------------------------------------------------------------------------
// Projection: z[n, 0:64] = (type==1 ? d_sim[n] @ W_d : m_sim[n] @ W_m)
// Masked concat GEMM with K = 512, fp32 WMMA 16x16x4.
// 1 block = one 16-row tile, 4 waves each own one 16-wide column tile.
// Weights staged through LDS in four K=128 tiles (kt 0,1 -> W_d; 2,3 -> W_m),
// using GLOBAL_LOAD_ASYNC_TO_LDS_B128 (ASYNCcnt) when the toolchain exposes
// it, else a batched register-staged copy.
// ---------------------------------------------------------------------------
__global__ __launch_bounds__(128)
void proj_wmma(const float* __restrict__ d_sim, const float* __restrict__ m_sim,
               const float* __restrict__ W_d,   const float* __restrict__ W_m,
               const int* __restrict__ node_type, float* __restrict__ z,
               int n_nodes) {
    __shared__ float lds_w[128 * FATT];          // 32 KB K-tile of W_cat

    const int tid  = threadIdx.x;                // 0..127
    const int wave = tid >> 5;                   // column tile 0..3
    const int lane = tid & 31;

    const int m    = lane & 15;                  // A row within tile
    const int kb   = (lane >> 4) * 2;            // A/B k sub-offset {0,2}
    const int row0 = blockIdx.x * 16;
    int row = row0 + m;
    if (row >= n_nodes) row = n_nodes - 1;       // clamp (keeps EXEC full)

    const int   t       = node_type[row];
    const float* arow_d = d_sim + (size_t)row * DSIM;
    const float* arow_m = m_sim + (size_t)row * DSIM;
    const int   ncol    = wave * 16 + m;         // B column (N = lane&15)

    v8f acc = {0.f, 0.f, 0.f, 0.f, 0.f, 0.f, 0.f, 0.f};

    float4* ldsv = (float4*)lds_w;

    for (int kt = 0; kt < 4; ++kt) {
        // stage 128x64 weight tile into LDS (2048 float4 / 128 threads)
        const float* Wsrc = (kt < 2) ? (W_d + (size_t)(kt * 128) * FATT)
                                     : (W_m + (size_t)((kt - 2) * 128) * FATT);
        const float4* wsv = (const float4*)Wsrc;
        __syncthreads();                          // previous-tile reads done
#ifdef USE_ASYNC_LDS
        #pragma unroll
        for (int i = 0; i < 16; ++i) {
            __builtin_amdgcn_global_load_async_to_lds_b128(
                (v4i*)(wsv + tid + i * 128),
                (lds_v4i_ptr)(ldsv + tid + i * 128),
                0, 0);
        }
        __builtin_amdgcn_s_wait_asynccnt(0);
#else
        float4 tmp[16];
        #pragma unroll
        for (int i = 0; i < 16; ++i) tmp[i] = wsv[tid + i * 128];
        #pragma unroll
        for (int i = 0; i < 16; ++i) ldsv[tid + i * 128] = tmp[i];
#endif
        __syncthreads();

        // A pointer for this K tile; mask = 1.0 if this tile's weight matrix
        // applies to this lane's row, else 0.0 (masked concat formulation).
        const float2* arow2 = (const float2*)((kt < 2) ? (arow_d + kt * 128)
                                                       : (arow_m + (kt - 2) * 128));
        const bool  use = (kt < 2) ? (t == 1) : (t != 1);
        const float mf  = use ? 1.0f : 0.0f;

        #pragma unroll 8
        for (int k = 0; k < 128; k += 4) {
            const int kk = k + kb;               // even
            const float2 av = arow2[kk >> 1];    // unconditional b64 load
            v2f a;
            a.x = av.x * mf;
            a.y = av.y * mf;
            v2f b;
            b.x = lds_w[kk * FATT + ncol];
            b.y = lds_w[(kk + 1) * FATT + ncol];
            // D = A(16x4,f32) * B(4x16,f32) + C ; fp32 WMMA
            acc = __builtin_amdgcn_wmma_f32_16x16x4_f32(
                false, a, false, b, (short)0, acc, false, false);
        }
    }

    // C/D layout: VGPR v -> M = 8*(lane>>4) + v, N = lane & 15
    const int nout  = wave * 16 + (lane & 15);
    const int mbase = (lane >> 4) * 8;
    #pragma unroll
    for (int v = 0; v < 8; ++v) {
        const int rr = row0 + mbase + v;
        if (rr < n_nodes) z[(size_t)rr * FATT + nout] = acc[v];
    }
}

// ---------------------------------------------------------------------------
// Exact float atomic max via int punning (works for mixed-sign values when
// the cell is initialized to -inf).
// ---------------------------------------------------------------------------
__device__ inline void atomic_max_f32(float* addr, float val) {
    if (val >= 0.0f)
        atomicMax((int*)addr, __float_as_int(val));
    else
        atomicMin((unsigned int*)addr, __float_as_uint(val));
}

// ---------------------------------------------------------------------------
// Edge logits: e = leaky_relu(<z[src], z[dst]>), fused segment-max into e_max.
// ---------------------------------------------------------------------------
__global__ __launch_bounds__(256)
void edge_score(const float* __restrict__ z, const int* __restrict__ src,
                const int* __restrict__ dst, float* __restrict__ e,
                float* __restrict__ e_max, int n_edges) {
    int i = blockIdx.x * blockDim.x + threadIdx.x;
    if (i >= n_edges) return;
    __builtin_prefetch(src + i + 4096, 0, 1);    // global_prefetch_b8
    __builtin_prefetch(dst + i + 4096, 0, 1);
    const int s = src[i], d = dst[i];
    const float4* zs = (const float4*)(z + (size_t)s * FATT);
    const float4* zd = (const float4*)(z + (size_t)d * FATT);
    float acc = 0.0f;
    #pragma unroll
    for (int j = 0; j < FATT / 4; ++j) {
        float4 a = zs[j], b = zd[j];
        acc += a.x * b.x + a.y * b.y + a.z * b.z + a.w * b.w;
    }
    acc = (acc > 0.0f) ? acc : SLOPE * acc;
    e[i] = acc;
    atomic_max_f32(&e_max[d], acc);
}

// ---------------------------------------------------------------------------
// ex = exp(e - e_max[dst]); denom[dst] += ex; hacc[dst] += ex * z[src].
// One wave (32 lanes) per edge; each lane scatters 2 features.
// ---------------------------------------------------------------------------
__global__ __launch_bounds__(256)
void edge_scatter(const float* __restrict__ z, const int* __restrict__ src,
                  const int* __restrict__ dst, const float* __restrict__ e,
                  const float* __restrict__ e_max, float* __restrict__ denom,
                  float* __restrict__ hacc, int n_edges) {
    const int w    = (blockIdx.x * blockDim.x + threadIdx.x) >> 5;  // edge id
    const int lane = threadIdx.x & 31;
    if (w >= n_edges) return;
    const int s = src[w], d = dst[w];
    const float ex = __expf(e[w] - e_max[d]);
    if (lane == 0) atomicAdd(&denom[d], ex);
    const float2 v = ((const float2*)(z + (size_t)s * FATT))[lane];
    float* hp = hacc + (size_t)d * FATT + lane * 2;
    atomicAdd(hp,     ex * v.x);
    atomicAdd(hp + 1, ex * v.y);
}

// ---------------------------------------------------------------------------
// out = elu(hacc / max(denom, 1e-16))
// ---------------------------------------------------------------------------
__global__ __launch_bounds__(256)
void finalize(const float* __restrict__ hacc, const float* __restrict__ denom,
              float* __restrict__ out, int n_nodes) {
    int i = blockIdx.x * blockDim.x + threadIdx.x;
    if (i >= n_nodes * FATT) return;
    const int node = i >> 6;
    const float h = hacc[i] / fmaxf(denom[node], 1e-16f);
    out[i] = (h > 0.0f) ? h : expm1f(h);
}

// ---------------------------------------------------------------------------
extern "C" void kernel_launch(void* const* d_in, const int* in_sizes, int n_in,
                              void* d_out, int out_size, void* d_ws, size_t ws_size,
                              hipStream_t stream) {
    (void)n_in; (void)out_size; (void)ws_size;
    const float* d_sim     = (const float*)d_in[0];
    const float* m_sim     = (const float*)d_in[1];
    const float* W_d       = (const float*)d_in[2];
    const float* W_m       = (const float*)d_in[3];
    const int*   node_type = (const int*)d_in[4];
    const int*   src       = (const int*)d_in[5];
    const int*   dst       = (const int*)d_in[6];
    const int N = in_sizes[4];
    const int E = in_sizes[5];

    float* ws    = (float*)d_ws;
    float* z     = ws;                          // N*64
    float* e     = z + (size_t)N * FATT;        // E
    float* e_max = e + (size_t)E;               // N
    float* denom = e_max + (size_t)N;           // N
    float* hacc  = denom + (size_t)N;           // N*64
    float* out   = (float*)d_out;

    const int nf = N * FATT;
    init_bufs   <<<(nf + 255) / 256, 256, 0, stream>>>(e_max, denom, hacc, N);
    proj_wmma   <<<(N + 15) / 16, 128, 0, stream>>>(d_sim, m_sim, W_d, W_m,
                                                    node_type, z, N);
    edge_score  <<<(E + 255) / 256, 256, 0, stream>>>(z, src, dst, e, e_max, E);
    edge_scatter<<<(E + 7) / 8, 256, 0, stream>>>(z, src, dst, e, e_max,
                                                  denom, hacc, E);
    finalize    <<<(nf + 255) / 256, 256, 0, stream>>>(hacc, denom, out, N);
}